// TMC_51539608000
// MI455X (gfx1250) — compile-verified
//
#include <hip/hip_runtime.h>
#include <cmath>

#define LOG2PI_F 1.8378770664093453f   // log(2*pi)
#define LOG2E_F  1.4426950408889634f
#define RES_F    0.5f                  // observed scalar (LIKE_STD = 1)

typedef __attribute__((ext_vector_type(2))) float v2f;
typedef __attribute__((ext_vector_type(8))) float v8f;

// raw v_exp_f32 (2^x), no denorm fixup: underflow flushes to 0, which is what LSE wants
__device__ __forceinline__ float exp2_raw(float x) {
  return __builtin_amdgcn_exp2f(x);
}

// ---------------------------------------------------------------------------
// Setup: zs[t,k] = mean_t + exp(log_stds[t]) * eps[t,k]
//        cc[t,k] = -log(sp) - 0.5*log2pi - log_q[t,k]   (per-column constant of M)
//        r[k]    = logpdf(zs[0,k]; 0, sp) - log_q[0,k]  (first factor)
// ---------------------------------------------------------------------------
__global__ void tmc_setup(const float* __restrict__ means,
                          const float* __restrict__ log_stds,
                          const float* __restrict__ eps,
                          float* __restrict__ zs, float* __restrict__ cc,
                          float* __restrict__ r, int T, int K,
                          float Tf, float logsp) {
  int idx = blockIdx.x * blockDim.x + threadIdx.x;
  if (idx >= T * K) return;
  int t = idx / K;
  float ls   = log_stds[t];
  float stdv = __expf(ls);
  float e    = eps[idx];
  float z    = means[t] + stdv * e;
  zs[idx] = z;
  float logq = -0.5f * e * e - ls - 0.5f * LOG2PI_F;
  cc[idx] = -logsp - 0.5f * LOG2PI_F - logq;
  if (t == 0) {
    r[idx] = -0.5f * z * z * Tf - logsp - 0.5f * LOG2PI_F - logq;
  }
}

// ---------------------------------------------------------------------------
// Prep: x = max_j r[j]; w[j] = exp(r[j]-x)  (w used only by the final kernel)
// Tile-major, lane-ready WMMA A operand for the next step (32 pairs per 16-row
// tile): pairs 0-15 = (negScale*z0^2 + (r-x)*log2e, -2*negScale*z0) for rows
// M=0..15 (lanes 0-15 supply A columns K=0,1); pairs 16-31 = (negScale, 0)
// (lanes 16-31 supply A columns K=2,3). Inner loop then needs zero selects.
// ---------------------------------------------------------------------------
__global__ void tmc_prep(const float* __restrict__ r, const float* __restrict__ z0row,
                         float* __restrict__ w, float* __restrict__ ap,
                         float* __restrict__ xbuf, int K, float negScale) {
  __shared__ float red[256];
  int tid = threadIdx.x;
  float m = -3.0e38f;
  for (int i = tid; i < K; i += 256) m = fmaxf(m, r[i]);
  red[tid] = m;
  __syncthreads();
  for (int s = 128; s > 0; s >>= 1) {
    if (tid < s) red[tid] = fmaxf(red[tid], red[tid + s]);
    __syncthreads();
  }
  float x = red[0];
  if (tid == 0) xbuf[0] = x;
  for (int i = tid; i < K; i += 256) {
    float rv = r[i];
    w[i] = __expf(rv - x);
    float z = z0row[i];
    int tile = i >> 4, mrow = i & 15;
    int base = tile * 64 + mrow * 2;        // floats: tile*32 pairs
    ap[base]          = fmaf(negScale * z, z, (rv - x) * LOG2E_F);
    ap[base + 1]      = -2.0f * negScale * z;
    ap[base + 32]     = negScale;           // hi-lane constant pair
    ap[base + 33]     = 0.0f;
  }
}

// ---------------------------------------------------------------------------
// Scan step: r_new[k] = x + cc[t,k] + log( sum_j 2^(D[j,k]) ) - log K
// D[j,k] = negScale*(z1_k - z0_j)^2 + log2(w_j) straight out of
// V_WMMA_F32_16X16X4_F32 (rank-3 factorization, weight folded into A[:,0]).
// Block = 16 columns, 4 waves stride over all K/16 row-tiles, 2 tiles/iter.
// ---------------------------------------------------------------------------
__global__ void __launch_bounds__(128) tmc_step(
    const float* __restrict__ zs, const float* __restrict__ cc,
    const float* __restrict__ ap, const float* __restrict__ xbuf,
    float* __restrict__ r, int t, int K, float logK) {
  extern __shared__ float smem[];
  float* s_a   = smem;            // 4*K floats (tile-major lane-ready A pairs)
  float* s_acc = smem + 4 * K;    // 16 floats

  const float* z1 = zs + (size_t)t * K;
  const int k0 = blockIdx.x * 16;

  for (int i = threadIdx.x; i < 4 * K; i += blockDim.x) s_a[i] = ap[i];
  if (threadIdx.x < 16) s_acc[threadIdx.x] = 0.0f;
  __syncthreads();

  const int  lane   = threadIdx.x & 31;
  const int  waveId = threadIdx.x >> 5;
  const int  n      = lane & 15;
  const bool hi     = lane >= 16;

  // B operand: lanes 0-15 carry rows K=0,1 -> (1, z1); lanes 16-31 rows K=2,3 -> (z1^2, 0)
  const float z1v = z1[k0 + n];
  v2f b;
  b.x = hi ? z1v * z1v : 1.0f;
  b.y = hi ? 0.0f      : z1v;

  const v2f* s_ap = (const v2f*)s_a;
  v8f c0 = {};
  float acc0 = 0.0f, acc1 = 0.0f, acc2 = 0.0f, acc3 = 0.0f;
  const int ntiles = K >> 4;

  auto body = [&](int jt) {
    v2f a = s_ap[(jt << 5) + lane];          // one ds_load_b64, WMMA-ready
    v8f d = __builtin_amdgcn_wmma_f32_16x16x4_f32(
        false, a, false, b, (short)0, c0, false, false);
    acc0 += exp2_raw(d[0]);
    acc1 += exp2_raw(d[1]);
    acc2 += exp2_raw(d[2]);
    acc3 += exp2_raw(d[3]);
    acc0 += exp2_raw(d[4]);
    acc1 += exp2_raw(d[5]);
    acc2 += exp2_raw(d[6]);
    acc3 += exp2_raw(d[7]);
  };

  int jt = waveId;
  for (; jt + 4 < ntiles; jt += 8) {         // 2 independent WMMA chains in flight
    body(jt);
    body(jt + 4);
  }
  if (jt < ntiles) body(jt);

  float acc = (acc0 + acc1) + (acc2 + acc3);

  // lanes l and l+16 hold the same column (rows 0-7 vs 8-15): SWAPX16 swizzle
  acc += __int_as_float(__builtin_amdgcn_ds_swizzle(__float_as_int(acc), 0x401f));
  if (!hi) atomicAdd(&s_acc[n], acc);        // combine 4 waves in LDS
  __syncthreads();

  if (threadIdx.x < 16) {
    int k = k0 + threadIdx.x;
    float S = s_acc[threadIdx.x];
    r[k] = xbuf[0] + cc[(size_t)t * K + k] + __logf(S) - logK;
  }
}

// ---------------------------------------------------------------------------
// Final: out = x + y + log( sum_j w[j]*exp(L[j]-y) ) - log K,  L[j] = logN(RES; z_last_j, 1)
// ---------------------------------------------------------------------------
__global__ void tmc_final(const float* __restrict__ zs, const float* __restrict__ w,
                          const float* __restrict__ xbuf, float* __restrict__ out,
                          int T, int K, float logK) {
  __shared__ float red[256];
  const float* zl = zs + (size_t)(T - 1) * K;
  int tid = threadIdx.x;
  float m = -3.0e38f;
  for (int i = tid; i < K; i += 256) {
    float d = RES_F - zl[i];
    float L = -0.5f * d * d - 0.5f * LOG2PI_F;
    m = fmaxf(m, L);
  }
  red[tid] = m;
  __syncthreads();
  for (int s = 128; s > 0; s >>= 1) {
    if (tid < s) red[tid] = fmaxf(red[tid], red[tid + s]);
    __syncthreads();
  }
  float y = red[0];
  __syncthreads();
  float s = 0.0f;
  for (int i = tid; i < K; i += 256) {
    float d = RES_F - zl[i];
    float L = -0.5f * d * d - 0.5f * LOG2PI_F;
    s += w[i] * __expf(L - y);
  }
  red[tid] = s;
  __syncthreads();
  for (int st = 128; st > 0; st >>= 1) {
    if (tid < st) red[tid] += red[tid + st];
    __syncthreads();
  }
  if (tid == 0) out[0] = xbuf[0] + y + __logf(red[0]) - logK;
}

// ---------------------------------------------------------------------------
extern "C" void kernel_launch(void* const* d_in, const int* in_sizes, int n_in,
                              void* d_out, int out_size, void* d_ws, size_t ws_size,
                              hipStream_t stream) {
  const float* means    = (const float*)d_in[0];
  const float* log_stds = (const float*)d_in[1];
  const float* eps      = (const float*)d_in[2];
  const int T = in_sizes[0];
  const int K = in_sizes[2] / T;

  const float Tf       = (float)T;
  const float logsp    = -0.5f * logf(Tf);          // log(sqrt(1/T))
  const float negScale = -0.5f * Tf * LOG2E_F;      // exp(-0.5*T*d2) == 2^(negScale*d2)
  const float logK     = logf((float)K);

  const size_t TK = (size_t)T * K;
  float* f  = (float*)d_ws;
  float* zs = f;             // T*K
  float* cc = zs + TK;       // T*K
  float* r  = cc + TK;       // K
  float* w  = r + K;         // K
  float* ap = w + K;         // 4*K (tile-major lane-ready A pairs)
  float* xb = ap + 4 * K;    // 1

  const int total = T * K;
  tmc_setup<<<(total + 255) / 256, 256, 0, stream>>>(means, log_stds, eps,
                                                     zs, cc, r, T, K, Tf, logsp);

  const size_t shbytes = (size_t)(4 * K + 16) * sizeof(float);
  for (int t = 1; t < T; ++t) {
    tmc_prep<<<1, 256, 0, stream>>>(r, zs + (size_t)(t - 1) * K, w, ap, xb, K, negScale);
    tmc_step<<<K / 16, 128, shbytes, stream>>>(zs, cc, ap, xb, r, t, K, logK);
  }
  tmc_prep<<<1, 256, 0, stream>>>(r, zs + (size_t)(T - 1) * K, w, ap, xb, K, negScale);
  tmc_final<<<1, 256, 0, stream>>>(zs, w, xb, (float*)d_out, T, K, logK);
}